// MultiHeadGraphAttentionLayer_51951924412552
// MI455X (gfx1250) — compile-verified
//
#include <hip/hip_runtime.h>
#include <stdint.h>

// dims: B=8, T=12 -> BT=96 ; N=512 ; Fin=64 ; H=8 ; Fp=8 ; C=H*Fp=64
#define BT   96
#define NN   512
#define FIN  64
#define CC   64      // H*Fp
#define HH   8
#define FP   8
#define PROW 516     // padded row stride for attention tile in LDS (516%64==4)

typedef float v2f __attribute__((ext_vector_type(2)));
typedef float v8f __attribute__((ext_vector_type(8)));

static __device__ __forceinline__ v8f wmma_f32_k4(v2f a, v2f b, v8f c) {
  // V_WMMA_F32_16X16X4_F32 : D = A(16x4 f32) * B(4x16 f32) + C(16x16 f32)
  return __builtin_amdgcn_wmma_f32_16x16x4_f32(
      /*neg_a=*/false, a, /*neg_b=*/false, b,
      /*c_mod=*/(short)0, c, /*reuse_a=*/false, /*reuse_b=*/false);
}

// Wave-relative LDS byte offset from a generic pointer to a __shared__ object
// (low 32 bits of a generic LDS address are the LDS byte address, ISA 10.2).
static __device__ __forceinline__ unsigned lds_off(const void* p) {
  return (unsigned)(uintptr_t)p;
}

// CDNA5 async global->LDS DMA, 16B per lane (GLOBAL_LOAD_ASYNC_TO_LDS_B128,
// GV mode: VDST = per-lane LDS byte offset, VADDR = 64-bit global address).
static __device__ __forceinline__ void async_g2lds_b128(unsigned ldsoff,
                                                        const void* gaddr) {
  asm volatile("global_load_async_to_lds_b128 %0, %1, off"
               :: "v"(ldsoff), "v"(gaddr)
               : "memory");
}
static __device__ __forceinline__ void wait_async0() {
  asm volatile("s_wait_asynccnt 0x0" ::: "memory");
}

// ---------------------------------------------------------------------------
// Kernel 1: h[bt][n][c] = x[bt][n][:] @ Wall[:][c] + bias[c]
//   Wall[i][c] = W[c>>3][i][c&7]  (heads stacked along columns)
//   One block per bt. 256 threads = 8 waves; 128 16x16 tiles -> 16 per wave.
// ---------------------------------------------------------------------------
__global__ __launch_bounds__(256) void gat_h_wmma(
    const float* __restrict__ x, const float* __restrict__ W,
    const float* __restrict__ Wb, float* __restrict__ hws) {
  __shared__ float wS[FIN * CC];  // Wall row-major [64][64], 16 KB

  const int tid = threadIdx.x;
  const int bt  = blockIdx.x;

  // Async-DMA stage of repacked W into LDS: 4-float chunks are contiguous in
  // both global (within one head-octet of a W row) and LDS.
  {
    const unsigned wbase = lds_off(&wS[0]);
    for (int m = tid; m < (FIN * CC) / 4; m += 256) {
      const int q0 = m * 4;
      const int i = q0 >> 6, c = q0 & 63;
      async_g2lds_b128(wbase + (unsigned)q0 * 4u,
                       W + (c >> 3) * (FIN * FP) + i * FP + (c & 7));
    }
    wait_async0();
  }
  __syncthreads();

  const int wave = tid >> 5, lane = tid & 31;
  const int rl = lane & 15;        // M (A) / N (B,C,D) within tile
  const int half = lane >> 4;      // selects K pair
  const float* xb = x + (size_t)bt * NN * FIN;
  float* hb = hws + (size_t)bt * NN * CC;

  for (int t = 0; t < 16; ++t) {
    const int tt = wave * 16 + t;
    const int rb = (tt >> 2) * 16;   // row tile base (n)
    const int cb = (tt & 3) * 16;    // col tile base (c)
    const int colN = cb + rl;
    const float bias = Wb[colN];
    v8f c = {bias, bias, bias, bias, bias, bias, bias, bias};
    const float* xrow = xb + (rb + rl) * FIN;
#pragma unroll
    for (int k = 0; k < FIN; k += 4) {
      const int kk = k + 2 * half;
      v2f a = *(const v2f*)(xrow + kk);                       // A: x[row][kk..kk+1]
      v2f b = {wS[kk * CC + colN], wS[(kk + 1) * CC + colN]}; // B: Wall[kk..][col]
      c = wmma_f32_k4(a, b, c);
    }
#pragma unroll
    for (int v = 0; v < 8; ++v) {
      const int m = v + 8 * half;  // D row per ISA layout
      hb[(rb + m) * CC + colN] = c[v];
    }
  }
}

// ---------------------------------------------------------------------------
// Kernel 2: fused masked softmax + attn@h per (bt, head, 16-row i-tile).
//   e[i,j] = lrelu(s_src[j] + s_dst[i] + ab[h]), -inf where adj==0,
//   softmax over j (unnormalized p kept in LDS; normalize after the matmul).
//   4 waves K-split the [16x512]x[512x16] fp32 WMMA matmul (cols>=8 are zero).
// ---------------------------------------------------------------------------
__global__ __launch_bounds__(128) void gat_attn_wmma(
    const float* __restrict__ adj, const float* __restrict__ a1,
    const float* __restrict__ a2, const float* __restrict__ ab,
    const float* __restrict__ hws, float* __restrict__ out) {
  __shared__ float pS[16 * PROW];   // unnormalized attention tile, 33 KB
  __shared__ float hS[NN * FP];     // h for this (bt, head), 16 KB
  __shared__ float sS[NN];          // s_src[j]
  __shared__ float sdS[16];         // s_dst[i] for the row tile
  __shared__ float redS[128];       // partial-reduce scratch (8 per row)
  __shared__ float rmaxS[16];
  __shared__ float rsumS[16];
  __shared__ float partS[4 * 256];  // per-wave partial D tiles

  const int tid  = threadIdx.x;
  const int bx   = blockIdx.x;
  const int itile = bx & 31;
  const int head  = (bx >> 5) & 7;
  const int bt    = bx >> 8;
  const int ib    = itile * 16;

  const float* hb = hws + (size_t)bt * NN * CC;

  if (tid < 16) __builtin_prefetch(adj + (ib + tid) * NN, 0, 3);

  // Async-DMA stage of h[:, head*8 .. +8] into LDS [j][f]: each row is two
  // contiguous 16B chunks in both global and LDS -> pure B128 async copies.
  {
    const unsigned hbase = lds_off(&hS[0]);
    for (int m = tid; m < (NN * FP) / 4; m += 128) {
      const int j = m >> 1, f4 = (m & 1) * 4;
      async_g2lds_b128(hbase + (unsigned)m * 16u,
                       hb + j * CC + head * FP + f4);
    }
    wait_async0();
  }

  float a1r[FP], a2r[FP];
#pragma unroll
  for (int f = 0; f < FP; ++f) {
    a1r[f] = a1[head * FP + f];
    a2r[f] = a2[head * FP + f];
  }
  const float abh = ab[head];
  __syncthreads();

  // s_src[j] = h[j] . a1 ; s_dst[i] = h[i] . a2  (from LDS)
  for (int q = tid; q < NN; q += 128) {
    float s = 0.f;
#pragma unroll
    for (int f = 0; f < FP; ++f) s += hS[q * FP + f] * a1r[f];
    sS[q] = s;
  }
  if (tid < 16) {
    float s = 0.f;
#pragma unroll
    for (int f = 0; f < FP; ++f) s += hS[(ib + tid) * FP + f] * a2r[f];
    sdS[tid] = s;
  }
  __syncthreads();

  // ---- Pass 1: masked row max (8 threads per row) ----
  const int r = tid >> 3, g = tid & 7;
  const float* adjrow = adj + (size_t)(ib + r) * NN;
  const float sd = sdS[r];
  float m = -__builtin_inff();
  for (int jj = 0; jj < 64; ++jj) {
    const int j = jj * 8 + g;
    const float av = adjrow[j];
    float e = sS[j] + sd + abh;
    e = e > 0.f ? e : 0.01f * e;
    if (av != 0.f) m = fmaxf(m, e);
  }
  redS[tid] = m;
  __syncthreads();
  if (tid < 16) {
    float mm = redS[tid * 8];
#pragma unroll
    for (int g2 = 1; g2 < 8; ++g2) mm = fmaxf(mm, redS[tid * 8 + g2]);
    rmaxS[tid] = mm;
  }
  __syncthreads();

  // ---- Pass 2: p = exp(e - max) (0 where masked) into LDS; row sums ----
  const float rm = rmaxS[r];
  float sacc = 0.f;
  for (int jj = 0; jj < 64; ++jj) {
    const int j = jj * 8 + g;
    const float av = adjrow[j];
    float e = sS[j] + sd + abh;
    e = e > 0.f ? e : 0.01f * e;
    const float p = (av != 0.f) ? __expf(e - rm) : 0.f;
    pS[r * PROW + j] = p;
    sacc += p;
  }
  redS[tid] = sacc;
  __syncthreads();
  if (tid < 16) {
    float ss = 0.f;
#pragma unroll
    for (int g2 = 0; g2 < 8; ++g2) ss += redS[tid * 8 + g2];
    rsumS[tid] = ss;
  }
  __syncthreads();

  // ---- WMMA: D(16x16) += p(16xK) * h(Kx16), K split over 4 waves ----
  const int wave = tid >> 5, lane = tid & 31;
  const int rowA = lane & 15;      // A row / B,D column index
  const int half = lane >> 4;      // K-pair select
  const int colB = rowA;
  const bool valid = colB < FP;    // cols 8..15 of B are zero padding
  v8f c = {0.f, 0.f, 0.f, 0.f, 0.f, 0.f, 0.f, 0.f};
  const int k0 = wave * 128;
  for (int k = 0; k < 128; k += 4) {
    const int kk = k0 + k + 2 * half;
    v2f a = *(const v2f*)(&pS[rowA * PROW + kk]);  // 8B aligned (kk even)
    float b0 = 0.f, b1 = 0.f;
    if (valid) {
      b0 = hS[kk * FP + colB];
      b1 = hS[(kk + 1) * FP + colB];
    }
    v2f b = {b0, b1};
    c = wmma_f32_k4(a, b, c);
  }
#pragma unroll
  for (int v = 0; v < 8; ++v) {
    const int mrow = v + 8 * half;
    partS[wave * 256 + mrow * 16 + colB] = c[v];
  }
  __syncthreads();

  // Combine 4 partials, normalize by row sum, write out[b,t,n,head*8+f].
  {
    const int mrow = tid >> 3, n = tid & 7;
    const float acc = partS[mrow * 16 + n] + partS[256 + mrow * 16 + n] +
                      partS[512 + mrow * 16 + n] + partS[768 + mrow * 16 + n];
    out[(size_t)bt * NN * CC + (ib + mrow) * CC + head * FP + n] =
        acc / rsumS[mrow];
  }
}

// ---------------------------------------------------------------------------
extern "C" void kernel_launch(void* const* d_in, const int* in_sizes, int n_in,
                              void* d_out, int out_size, void* d_ws,
                              size_t ws_size, hipStream_t stream) {
  (void)in_sizes; (void)n_in; (void)out_size; (void)ws_size;
  const float* x   = (const float*)d_in[0];
  const float* adj = (const float*)d_in[1];
  const float* W   = (const float*)d_in[2];
  const float* Wb  = (const float*)d_in[3];
  const float* a1  = (const float*)d_in[4];
  const float* a2  = (const float*)d_in[5];
  const float* ab  = (const float*)d_in[6];
  float* out = (float*)d_out;
  float* hws = (float*)d_ws;  // h intermediate: 96*512*64 floats (12.6 MB)

  gat_h_wmma<<<BT, 256, 0, stream>>>(x, W, Wb, hws);
  gat_attn_wmma<<<BT * HH * (NN / 16), 128, 0, stream>>>(adj, a1, a2, ab, hws,
                                                         out);
}